// InpaintingLoss_79001628443370
// MI455X (gfx1250) — compile-verified
//
#include <hip/hip_runtime.h>

typedef __attribute__((ext_vector_type(2))) float v2f;
typedef __attribute__((ext_vector_type(8))) float v8f;

#define B_SZ   8
#define NL_SZ  256
#define NP_SZ  16384
#define DL_SZ  10
#define DP_SZ  4
#define DELTA_C 0.01f
#define L_BIND_C 1.0f
#define L_MREG_C 0.1f

// ---------------------------------------------------------------------------
// ws[0] = sum(mse * mask)   ws[1] = sum(mask)   ws[2] = sum(mdiff * mask)
// ws[3] = total binding energy (atomic)
// ---------------------------------------------------------------------------

__global__ void init_ws_kernel(float* ws) {
    if (threadIdx.x < 4) ws[threadIdx.x] = 0.0f;
}

// One block: masked denoise + mreg reductions over B*NL = 2048 atoms.
__global__ __launch_bounds__(256)
void masked_losses_kernel(const float* __restrict__ pn,
                          const float* __restrict__ tc,
                          const float* __restrict__ sc,
                          const float* __restrict__ mask,
                          float* __restrict__ ws) {
    __shared__ float s0[256], s1[256], s2[256];
    const int tid = threadIdx.x;
    float a0 = 0.f, a1 = 0.f, a2 = 0.f;
    for (int i = tid; i < B_SZ * NL_SZ; i += 256) {
        float m = mask[i];
        float mse = 0.f, md = 0.f;
#pragma unroll
        for (int d = 0; d < 3; ++d) {
            float t  = tc[i * 3 + d];
            float e1 = pn[i * 3 + d] - t;
            float e2 = sc[i * 3 + d] - t;
            mse += e1 * e1;
            md  += e2 * e2;
        }
        a0 += m * mse;
        a1 += m;
        a2 += m * md;
    }
    s0[tid] = a0; s1[tid] = a1; s2[tid] = a2;
    __syncthreads();
    for (int off = 128; off > 0; off >>= 1) {
        if (tid < off) {
            s0[tid] += s0[tid + off];
            s1[tid] += s1[tid + off];
            s2[tid] += s2[tid + off];
        }
        __syncthreads();
    }
    if (tid == 0) { ws[0] = s0[0]; ws[1] = s1[0]; ws[2] = s2[0]; }
}

// ---------------------------------------------------------------------------
// Binding energy via V_WMMA_F32_16X16X4_F32.
//   A(m,0..2) = x coords, A(m,3) = 1
//   B(0..2,n) = -2 * p coords, B(3,n) = |p_n|^2   (per-tile)
//   C(m,n)    = |x_m|^2                            (loop-invariant per lane)
// => D = |x|^2 - 2 x.p + |p|^2 = squared distances, one 16x16 tile per WMMA.
// Grid: B * (NL/16) blocks, 8 waves/block; each wave strides the 1024
// protein column tiles. EXEC is all-ones everywhere the WMMA executes.
// ---------------------------------------------------------------------------
__global__ __launch_bounds__(256)
void bind_energy_kernel(const float* __restrict__ x,        // target_coords2 (B,NL,3)
                        const float* __restrict__ lig_feat, // (B,NL,DL)
                        const float* __restrict__ prot,     // (B,NP,3)
                        const float* __restrict__ prot_feat,// (B,NP,DP)
                        const float* __restrict__ lig_tab,  // (DL)
                        const float* __restrict__ prot_tab, // (DP)
                        float* __restrict__ ws) {
    const int b    = blockIdx.x >> 4;   // batch
    const int rt   = blockIdx.x & 15;   // row tile within NL
    const int lane = threadIdx.x & 31;
    const int wave = threadIdx.x >> 5;
    const int hi   = lane >> 4;         // 0: lanes 0-15, 1: lanes 16-31
    const int nl   = lane & 15;         // local row/col index within tile

    // Charge tables into registers.
    float ltab[DL_SZ];
#pragma unroll
    for (int i = 0; i < DL_SZ; ++i) ltab[i] = lig_tab[i];
    float ptab[DP_SZ];
#pragma unroll
    for (int i = 0; i < DP_SZ; ++i) ptab[i] = prot_tab[i];

    // --- A matrix (16x4 f32): fixed for the wave; row m = rt*16 + nl ---
    const int m0 = rt * 16 + nl;
    const float* xb = x + ((size_t)b * NL_SZ + m0) * 3;
    float x0 = xb[0], x1 = xb[1], x2 = xb[2];
    v2f amat;
    amat.x = hi ? x2   : x0;   // K=2 : K=0
    amat.y = hi ? 1.0f : x1;   // K=3 : K=1

    // Loop-invariant C accumulator seed (|x_m|^2) and ligand charges for the
    // 8 rows this lane's accumulators cover (C/D layout: VGPR i -> M = i+8*hi).
    v8f cseed;
    float qlv[8];
#pragma unroll
    for (int i = 0; i < 8; ++i) {
        int mr = rt * 16 + i + 8 * hi;
        const float* xr = x + ((size_t)b * NL_SZ + mr) * 3;
        float r0 = xr[0], r1 = xr[1], r2 = xr[2];
        cseed[i] = r0 * r0 + r1 * r1 + r2 * r2;

        const float* lf = lig_feat + ((size_t)b * NL_SZ + mr) * DL_SZ;
        float best = lf[0];
        int bi = 0;
#pragma unroll
        for (int j = 1; j < DL_SZ; ++j) {
            float v = lf[j];
            if (v > best) { best = v; bi = j; }
        }
        qlv[i] = ltab[bi];
    }

    float acc = 0.f;
    const int NCT = NP_SZ / 16;          // 1024 column tiles
    for (int ct = wave; ct < NCT; ct += 8) {
        const int n = ct * 16 + nl;      // protein atom for this lane's column
        const float* pb = prot + ((size_t)b * NP_SZ + n) * 3;
        float p0 = pb[0], p1 = pb[1], p2 = pb[2];
        float psq = p0 * p0 + p1 * p1 + p2 * p2;

        v2f bmat;
        bmat.x = hi ? (-2.f * p2) : (-2.f * p0);  // K=2 : K=0
        bmat.y = hi ? psq         : (-2.f * p1);  // K=3 : K=1

        // D = A*B + C -> 16x16 tile of squared distances.
        v8f d2 = __builtin_amdgcn_wmma_f32_16x16x4_f32(
            /*neg_a=*/false, amat, /*neg_b=*/false, bmat,
            /*c_mod=*/(short)0, cseed, /*reuse_a=*/false, /*reuse_b=*/false);

        // Protein charge for this column (rows are 16B, vector load).
        const float4 pf = *(const float4*)(prot_feat + ((size_t)b * NP_SZ + n) * DP_SZ);
        float qp = ptab[0];
        float bp = pf.x;
        if (pf.y > bp) { bp = pf.y; qp = ptab[1]; }
        if (pf.z > bp) { bp = pf.z; qp = ptab[2]; }
        if (pf.w > bp) { bp = pf.w; qp = ptab[3]; }

#pragma unroll
        for (int i = 0; i < 8; ++i) {
            // Raw TRANS ops (v_sqrt_f32 / v_rcp_f32): co-execute with WMMA,
            // skip the precise-mode fixup sequences.
            float s    = __builtin_amdgcn_sqrtf(fmaxf(d2[i], 0.f));
            float inv  = __builtin_amdgcn_rcpf(s + DELTA_C);
            float inv2 = inv * inv;
            float inv6 = inv2 * inv2 * inv2;
            // EPS = A_C = B_C = 1
            acc += qlv[i] * qp * inv + inv6 * inv6 - inv6;
        }
    }

    // Wave32 tree reduction, then one atomic per wave.
#pragma unroll
    for (int off = 16; off > 0; off >>= 1)
        acc += __shfl_xor(acc, off, 32);
    if (lane == 0) atomicAdd(&ws[3], acc);
}

__global__ void finalize_kernel(const float* __restrict__ ws,
                                float* __restrict__ out) {
    if (threadIdx.x == 0 && blockIdx.x == 0) {
        float msum = ws[1];
        float loss_denoise = ws[0] / msum;
        float loss_bind    = ws[3] / (float)B_SZ;
        float loss_mreg    = ws[2] / msum;
        out[0] = loss_denoise + L_BIND_C * loss_bind + L_MREG_C * loss_mreg;
    }
}

extern "C" void kernel_launch(void* const* d_in, const int* in_sizes, int n_in,
                              void* d_out, int out_size, void* d_ws, size_t ws_size,
                              hipStream_t stream) {
    const float* pn    = (const float*)d_in[0]; // predicted_noise
    const float* tc    = (const float*)d_in[1]; // target_coords
    const float* sc    = (const float*)d_in[2]; // scaffold_coords
    const float* tc2   = (const float*)d_in[3]; // target_coords2
    const float* tfeat = (const float*)d_in[4]; // target_features
    const float* mask  = (const float*)d_in[5]; // mask
    const float* prot  = (const float*)d_in[6]; // protein_coords
    const float* pfeat = (const float*)d_in[7]; // protein_features
    const float* ltab  = (const float*)d_in[8]; // lig_charge_table
    const float* ptab  = (const float*)d_in[9]; // prot_charge_table
    float* out = (float*)d_out;
    float* ws  = (float*)d_ws;

    init_ws_kernel<<<1, 32, 0, stream>>>(ws);
    masked_losses_kernel<<<1, 256, 0, stream>>>(pn, tc, sc, mask, ws);
    bind_energy_kernel<<<B_SZ * (NL_SZ / 16), 256, 0, stream>>>(
        tc2, tfeat, prot, pfeat, ltab, ptab, ws);
    finalize_kernel<<<1, 32, 0, stream>>>(ws, out);
}